// Curvature_71579924955438
// MI455X (gfx1250) — compile-verified
//
#include <hip/hip_runtime.h>
#include <hip/hip_bf16.h>
#include <math.h>

#define GRID   32
#define NCELLS (GRID * GRID * GRID)   // 32768
#define NT     48
#define MAXTRI 4
#define G1     33                      // GRID+1
#define EPSF   1e-8f
#define CELLS_PER_BLK 128

typedef unsigned int u32x4 __attribute__((ext_vector_type(4)));
typedef int          i32x8 __attribute__((ext_vector_type(8)));
typedef int          i32x4 __attribute__((ext_vector_type(4)));
typedef float        v2f   __attribute__((ext_vector_type(2)));
typedef float        v8f   __attribute__((ext_vector_type(8)));

// ---------------------------------------------------------------------------
// Compile-time reproduction of numpy.random.RandomState(0) draws:
//   TRI_TABLE = randint(0,12,(48,4,3))  -> masked rejection, mask=15, rng=11
//   _ntri     = randint(1,5,(48,))      -> mask=3, rng=3 (never rejects), off=1
// Legacy MT19937 seeding: key[p]=seed; seed = 1812433253*(seed^(seed>>30))+p+1
// ---------------------------------------------------------------------------
struct MTState { unsigned int mt[624]; int idx; };

constexpr unsigned int mt_next(MTState &s) {
  if (s.idx >= 624) {
    for (int i = 0; i < 624; ++i) {
      unsigned int y = (s.mt[i] & 0x80000000u) | (s.mt[(i + 1) % 624] & 0x7fffffffu);
      unsigned int v = s.mt[(i + 397) % 624] ^ (y >> 1);
      if (y & 1u) v ^= 0x9908b0dfu;
      s.mt[i] = v;
    }
    s.idx = 0;
  }
  unsigned int y = s.mt[s.idx++];
  y ^= y >> 11;
  y ^= (y << 7) & 0x9d2c5680u;
  y ^= (y << 15) & 0xefc60000u;
  y ^= y >> 18;
  return y;
}

struct Tables {
  int tri[NT][MAXTRI][3];
  int ntri[NT];
};

constexpr Tables make_tables() {
  Tables tb{};
  MTState s{};
  unsigned int seed = 0u;
  for (int i = 0; i < 624; ++i) {
    s.mt[i] = seed;
    seed = 1812433253u * (seed ^ (seed >> 30)) + (unsigned int)i + 1u;
  }
  s.idx = 624;
  for (int t = 0; t < NT; ++t)
    for (int p = 0; p < MAXTRI; ++p)
      for (int v = 0; v < 3; ++v) {
        unsigned int val = 0u;
        do { val = mt_next(s) & 15u; } while (val > 11u);
        tb.tri[t][p][v] = (int)val;
      }
  for (int t = 0; t < NT; ++t)
    tb.ntri[t] = 1 + (int)(mt_next(s) & 3u);
  return tb;
}

__constant__ Tables g_tab = make_tables();

// EDGES table: (dx, dy, dz, axis) — compile-time so the vertex loop folds.
static constexpr int kEdge[12][4] = {
    {0, 0, 0, 0}, {1, 0, 0, 1}, {0, 1, 0, 0}, {0, 0, 0, 1},
    {0, 0, 1, 0}, {1, 0, 1, 1}, {0, 1, 1, 0}, {0, 0, 1, 1},
    {0, 0, 0, 2}, {1, 0, 0, 2}, {1, 1, 0, 2}, {0, 1, 0, 2}};

// ---------------------------------------------------------------------------
// Kernel 0: zero the scalar accumulator (harness poisons d_out).
// ---------------------------------------------------------------------------
__global__ void init_kernel(float* __restrict__ out) { out[0] = 0.0f; }

// ---------------------------------------------------------------------------
// Kernel 1: per-cell pass.
//  - TDM DMA of the topology tile [128 x 48] f32 into LDS (tensor_load_to_lds)
//  - 12 edge vertices -> LDS
//  - per-topology normals / within / mean-normal, fused reductions
//  - P[n] (topology row sums) via v_wmma_f32_16x16x4_f32 with B = ones
// ---------------------------------------------------------------------------
struct SharedBlk {
  float topo[CELLS_PER_BLK][NT];       // MUST be first: TDM writes LDS offset 0
  float vert[CELLS_PER_BLK][12][3];
  float red[CELLS_PER_BLK];
};

__global__ __launch_bounds__(CELLS_PER_BLK) void cell_kernel(
    const float* __restrict__ offset, const float* __restrict__ topology,
    float* __restrict__ ex, float* __restrict__ ey, float* __restrict__ ez,
    float* __restrict__ sv, float* __restrict__ pv, float* __restrict__ out) {
  __shared__ SharedBlk sh;
  const int tx  = (int)threadIdx.x;
  const int blk = (int)blockIdx.x;
  const int n   = blk * CELLS_PER_BLK + tx;

  // ---- Issue the TDM descriptor from wave 0 (D# per CDNA5 ISA) ----
  if (tx < 32) {
    unsigned long long ga =
        (unsigned long long)(topology + (size_t)blk * CELLS_PER_BLK * NT);
    u32x4 g0;
    g0[0] = 1u;                                        // count=1, user mode
    g0[1] = 0u;                                        // lds_addr = 0 (sh.topo)
    g0[2] = (unsigned int)(ga & 0xffffffffu);          // global_addr[31:0]
    g0[3] = (unsigned int)((ga >> 32) & 0x1ffffffu) |  // global_addr[56:32]
            (2u << 30);                                // type = 2 ("image")
    i32x8 g1;
    g1[0] = (int)(2u << 16);                 // data_size = 2 (4 bytes)
    g1[1] = (int)((unsigned)NT << 16);       // tensor_dim0[15:0] = 48
    g1[2] = (int)((unsigned)CELLS_PER_BLK << 16);  // tensor_dim1[15:0] = 128
    g1[3] = (int)((unsigned)NT << 16);       // tile_dim0 = 48
    g1[4] = CELLS_PER_BLK;                   // tile_dim1 = 128
    g1[5] = NT;                              // tensor_dim0_stride = 48
    g1[6] = 0;
    g1[7] = 0;
    i32x4 g2 = {0, 0, 0, 0};
    i32x4 g3 = {0, 0, 0, 0};
    i32x8 g4 = {0, 0, 0, 0, 0, 0, 0, 0};     // extra group (clang-23 6-arg form)
    __builtin_amdgcn_tensor_load_to_lds(g0, g1, g2, g3, g4, 0);
  }

  // ---- Edge vertices (overlaps with the DMA) ----
  const int k = n & 31, j = (n >> 5) & 31, i = n >> 10;
#pragma unroll
  for (int e = 0; e < 12; ++e) {
    const int a  = kEdge[e][3];
    const int gi = i + kEdge[e][0];
    const int gj = j + kEdge[e][1];
    const int gk = k + kEdge[e][2];
    const float disp = offset[(((size_t)a * G1 + gi) * G1 + gj) * G1 + gk];
    float vx = (float)gi, vy = (float)gj, vz = (float)gk;
    const float d = 0.5f + disp;
    if (a == 0) vx += d; else if (a == 1) vy += d; else vz += d;
    sh.vert[tx][e][0] = vx;
    sh.vert[tx][e][1] = vy;
    sh.vert[tx][e][2] = vz;
  }
  if (tx < 32) __builtin_amdgcn_s_wait_tensorcnt(0);
  __syncthreads();

  // ---- Per-topology normals and fused reductions ----
  float e0 = 0.0f, e1 = 0.0f, e2 = 0.0f, sacc = 0.0f, lw = 0.0f;
  for (int t = 0; t < NT; ++t) {
    float nx[4], ny[4], nz[4];
#pragma unroll
    for (int p = 0; p < MAXTRI; ++p) {
      const int ia = g_tab.tri[t][p][0];
      const int ib = g_tab.tri[t][p][1];
      const int ic = g_tab.tri[t][p][2];
      const float ux = sh.vert[tx][ib][0] - sh.vert[tx][ia][0];
      const float uy = sh.vert[tx][ib][1] - sh.vert[tx][ia][1];
      const float uz = sh.vert[tx][ib][2] - sh.vert[tx][ia][2];
      const float wx = sh.vert[tx][ic][0] - sh.vert[tx][ia][0];
      const float wy = sh.vert[tx][ic][1] - sh.vert[tx][ia][1];
      const float wz = sh.vert[tx][ic][2] - sh.vert[tx][ia][2];
      const float cx = uy * wz - uz * wy;
      const float cy = uz * wx - ux * wz;
      const float cz = ux * wy - uy * wx;
      const float nrm = sqrtf(cx * cx + cy * cy + cz * cz);
      const float inv = 1.0f / (nrm + EPSF);
      nx[p] = cx * inv; ny[p] = cy * inv; nz[p] = cz * inv;
    }
    const int ntri = g_tab.ntri[t];
    float w = 0.0f;
#pragma unroll
    for (int p = 0; p < MAXTRI - 1; ++p) {
      if (p + 1 < ntri) {
        const float dx = nx[p + 1] - nx[p];
        const float dy = ny[p + 1] - ny[p];
        const float dz = nz[p + 1] - nz[p];
        w += dx * dx + dy * dy + dz * dz;
      }
    }
    float mx = 0.0f, my = 0.0f, mz = 0.0f;
#pragma unroll
    for (int p = 0; p < MAXTRI; ++p)
      if (p < ntri) { mx += nx[p]; my += ny[p]; mz += nz[p]; }
    const float invc = 1.0f / (float)ntri;
    mx *= invc; my *= invc; mz *= invc;

    const float tp = sh.topo[tx][t];
    lw   += tp * w;
    e0   += tp * mx;
    e1   += tp * my;
    e2   += tp * mz;
    sacc += tp * (mx * mx + my * my + mz * mz);
  }
  ex[n] = e0; ey[n] = e1; ez[n] = e2; sv[n] = sacc;

  // ---- P[n] = sum_t topology[n,t]: WMMA row-sums, 16 cells per tile ----
  // A (16x4 f32): lanes 0-15 hold {K=0,K=1}, lanes 16-31 hold {K=2,K=3};
  // B = all-ones (layout invariant). D columns are identical = row sums.
  {
    const int lane = tx & 31;
    const int mrow = lane & 15;
    const int kcol = (lane >> 4) << 1;  // 0 or 2
    const int wvid = tx >> 5;
    const v2f bones = {1.0f, 1.0f};
#pragma unroll
    for (int gsub = 0; gsub < 2; ++gsub) {
      const int grp = wvid * 2 + gsub;  // 16-cell group within the block
      v8f acc = {0.f, 0.f, 0.f, 0.f, 0.f, 0.f, 0.f, 0.f};
#pragma unroll
      for (int c = 0; c < NT / 4; ++c) {
        v2f av;
        av.x = sh.topo[grp * 16 + mrow][c * 4 + kcol];
        av.y = sh.topo[grp * 16 + mrow][c * 4 + kcol + 1];
        acc = __builtin_amdgcn_wmma_f32_16x16x4_f32(
            false, av, false, bones, (short)0, acc, false, false);
      }
      // C/D layout: VGPR r holds M=r (lanes 0-15) and M=8+r (lanes 16-31).
      if (mrow == 0) {  // lanes 0 and 16 write their 8 rows
        const int mb = (lane >> 4) * 8;
#pragma unroll
        for (int r = 0; r < 8; ++r)
          pv[blk * CELLS_PER_BLK + grp * 16 + mb + r] = acc[r];
      }
    }
  }

  // ---- block-reduce loss_within, one atomic per block ----
  sh.red[tx] = lw;
  __syncthreads();
  for (int off = CELLS_PER_BLK / 2; off > 0; off >>= 1) {
    if (tx < off) sh.red[tx] += sh.red[tx + off];
    __syncthreads();
  }
  if (tx == 0) atomicAdd(out, sh.red[0]);
}

// ---------------------------------------------------------------------------
// Kernel 2: neighbor stencil  sum over 3 axes of s_l*P_r + s_r*P_l - 2 e_l.e_r
// ---------------------------------------------------------------------------
__global__ __launch_bounds__(256) void cross_kernel(
    const float* __restrict__ ex, const float* __restrict__ ey,
    const float* __restrict__ ez, const float* __restrict__ sv,
    const float* __restrict__ pv, float* __restrict__ out) {
  __shared__ float red[256];
  const int tx = (int)threadIdx.x;
  const int n  = (int)blockIdx.x * 256 + tx;
  const int k = n & 31, j = (n >> 5) & 31, i = n >> 10;

  const float el0 = ex[n], el1 = ey[n], el2 = ez[n];
  const float sl = sv[n], pl = pv[n];
  float acc = 0.0f;
  if (k < GRID - 1) {
    const int r = n + 1;
    acc += sl * pv[r] + sv[r] * pl -
           2.0f * (el0 * ex[r] + el1 * ey[r] + el2 * ez[r]);
  }
  if (j < GRID - 1) {
    const int r = n + GRID;
    acc += sl * pv[r] + sv[r] * pl -
           2.0f * (el0 * ex[r] + el1 * ey[r] + el2 * ez[r]);
  }
  if (i < GRID - 1) {
    const int r = n + GRID * GRID;
    acc += sl * pv[r] + sv[r] * pl -
           2.0f * (el0 * ex[r] + el1 * ey[r] + el2 * ez[r]);
  }

  red[tx] = acc;
  __syncthreads();
  for (int off = 128; off > 0; off >>= 1) {
    if (tx < off) red[tx] += red[tx + off];
    __syncthreads();
  }
  if (tx == 0) atomicAdd(out, red[0]);
}

// ---------------------------------------------------------------------------
extern "C" void kernel_launch(void* const* d_in, const int* in_sizes, int n_in,
                              void* d_out, int out_size, void* d_ws,
                              size_t ws_size, hipStream_t stream) {
  (void)in_sizes; (void)n_in; (void)out_size; (void)ws_size;
  const float* offset   = (const float*)d_in[0];  // [3,33,33,33]
  const float* topology = (const float*)d_in[1];  // [32768,48]
  float* out = (float*)d_out;                     // [1]

  float* ws = (float*)d_ws;                       // 5 * 32768 floats = 640 KB
  float* ex = ws;
  float* ey = ws + NCELLS;
  float* ez = ws + 2 * NCELLS;
  float* sv = ws + 3 * NCELLS;
  float* pv = ws + 4 * NCELLS;

  hipLaunchKernelGGL(init_kernel, dim3(1), dim3(1), 0, stream, out);
  hipLaunchKernelGGL(cell_kernel, dim3(NCELLS / CELLS_PER_BLK),
                     dim3(CELLS_PER_BLK), 0, stream, offset, topology, ex, ey,
                     ez, sv, pv, out);
  hipLaunchKernelGGL(cross_kernel, dim3(NCELLS / 256), dim3(256), 0, stream,
                     ex, ey, ez, sv, pv, out);
}